// SAGE_73675868995819
// MI455X (gfx1250) — compile-verified
//
#include <hip/hip_runtime.h>
#include <math.h>

// Problem constants (match reference)
#define N_NODES 100000
#define E_EDGES 800000
#define F_IN    100
#define HID     128
#define NCLS    64
#define EPSN    1e-12f

typedef __attribute__((ext_vector_type(2))) float v2f;
typedef __attribute__((ext_vector_type(8))) float v8f;

// ---------------------------------------------------------------------------
// 1) CSR row pointers from sorted edge_rows: row_ptr[i] = lower_bound(rows, i)
// ---------------------------------------------------------------------------
__global__ void build_row_ptr_kernel(const int* __restrict__ rows,
                                     int* __restrict__ row_ptr) {
  int i = blockIdx.x * blockDim.x + threadIdx.x;
  if (i > N_NODES) return;
  int lo = 0, hi = E_EDGES;
  while (lo < hi) {
    int mid = (lo + hi) >> 1;
    if (rows[mid] < i) lo = mid + 1; else hi = mid;
  }
  row_ptr[i] = lo;
}

// ---------------------------------------------------------------------------
// 2) CSR SpMM: out[i,f] = sum_e vals[e] * x[cols[e], f]   (block per node)
// ---------------------------------------------------------------------------
template <int F>
__global__ __launch_bounds__(128)
void spmm_csr_kernel(const int* __restrict__ row_ptr,
                     const int* __restrict__ cols,
                     const float* __restrict__ vals,
                     const float* __restrict__ x,
                     float* __restrict__ out) {
  const int node = blockIdx.x;
  const int f = threadIdx.x;
  if (f >= F) return;
  const int e0 = row_ptr[node], e1 = row_ptr[node + 1];
  float acc = 0.f;
  for (int e = e0; e < e1; ++e) {
    acc = fmaf(vals[e], x[(long)cols[e] * F + f], acc);
  }
  out[(long)node * F + f] = acc;
}

// ---------------------------------------------------------------------------
// 3) Layer 1: h = relu(l1norm(ax@Wl.T + x@Wr.T + (bl+br)))
//    Block = 256 threads = 8 waves; block computes rows [16b,16b+16) x 128 cols.
//    Wave w computes the 16x16 tile at columns [16w,16w+16) via WMMA f32.
// ---------------------------------------------------------------------------
__global__ __launch_bounds__(256)
void layer1_kernel(const float* __restrict__ ax,   // [N, F_IN]
                   const float* __restrict__ x,    // [N, F_IN]
                   const float* __restrict__ Wl,   // [HID, F_IN] row-major
                   const float* __restrict__ bl,   // [HID]
                   const float* __restrict__ Wr,   // [HID, F_IN]
                   const float* __restrict__ br,   // [HID]
                   float* __restrict__ h) {        // [N, HID]
  __shared__ float sA[16 * F_IN];
  __shared__ float sX[16 * F_IN];
  __shared__ float rowsum[16];

  const int row0 = blockIdx.x * 16;
  const int tid  = threadIdx.x;
  const int wave = tid >> 5;
  const int lane = tid & 31;

  // Stage 16 contiguous rows of ax and x into LDS (coalesced).
  for (int idx = tid; idx < 16 * F_IN; idx += 256) {
    sA[idx] = ax[(long)row0 * F_IN + idx];
    sX[idx] = x [(long)row0 * F_IN + idx];
  }
  if (tid < 16) rowsum[tid] = 0.f;
  __syncthreads();

  // WMMA f32 16x16x4 fragment indices (ISA 7.12.2):
  //   A: lane -> M = lane&15 ; K pair at kh = (lane>>4)*2
  //   B: lane -> N = lane&15 ; same K striping
  const int m    = lane & 15;
  const int kh   = (lane >> 4) * 2;
  const int half = lane >> 4;
  const int n    = wave * 16 + (lane & 15);   // output column / weight row

  v8f c = {};
  #pragma unroll 5
  for (int k = 0; k < F_IN; k += 4) {
    v2f a, b;
    // left GEMM: ax @ Wl.T  (B[k][n] = Wl[n*F_IN + k])
    a.x = sA[m * F_IN + k + kh];
    a.y = sA[m * F_IN + k + kh + 1];
    b.x = Wl[n * F_IN + k + kh];
    b.y = Wl[n * F_IN + k + kh + 1];
    c = __builtin_amdgcn_wmma_f32_16x16x4_f32(false, a, false, b,
                                              (short)0, c, false, false);
    // right GEMM: x @ Wr.T (accumulate into same C)
    a.x = sX[m * F_IN + k + kh];
    a.y = sX[m * F_IN + k + kh + 1];
    b.x = Wr[n * F_IN + k + kh];
    b.y = Wr[n * F_IN + k + kh + 1];
    c = __builtin_amdgcn_wmma_f32_16x16x4_f32(false, a, false, b,
                                              (short)0, c, false, false);
  }

  const float bias = bl[n] + br[n];
  float t[8];
  // L1 row sums: reduce |t| over the 16 lanes holding each row, then ds-atomic
  // across the 8 waves (each wave owns 16 of the 128 columns).
  #pragma unroll
  for (int r = 0; r < 8; ++r) {
    t[r] = c[r] + bias;
    float s = fabsf(t[r]);
    s += __shfl_xor(s, 1);
    s += __shfl_xor(s, 2);
    s += __shfl_xor(s, 4);
    s += __shfl_xor(s, 8);
    if ((lane & 15) == 0) atomicAdd(&rowsum[r + half * 8], s);
  }
  __syncthreads();

  #pragma unroll
  for (int r = 0; r < 8; ++r) {
    const int rowm = r + half * 8;
    const float inv = 1.f / fmaxf(rowsum[rowm], EPSN);
    const float v = fmaxf(t[r] * inv, 0.f);   // normalize then ReLU
    h[(long)(row0 + rowm) * HID + n] = v;
  }
}

// ---------------------------------------------------------------------------
// 5) Layer 2: out = log_softmax(ah@Wl2.T + h@Wr2.T + (bl2+br2))
//    Block = 128 threads = 4 waves; rows [16b,16b+16) x 64 cols.
// ---------------------------------------------------------------------------
__global__ __launch_bounds__(128)
void layer2_kernel(const float* __restrict__ ah,   // [N, HID]
                   const float* __restrict__ h,    // [N, HID]
                   const float* __restrict__ Wl,   // [NCLS, HID]
                   const float* __restrict__ bl,   // [NCLS]
                   const float* __restrict__ Wr,   // [NCLS, HID]
                   const float* __restrict__ br,   // [NCLS]
                   float* __restrict__ out) {      // [N, NCLS]
  __shared__ float sA[16 * HID];
  __shared__ float sH[16 * HID];
  __shared__ float pmax[4][16];
  __shared__ float psum[4][16];

  const int row0 = blockIdx.x * 16;
  const int tid  = threadIdx.x;
  const int wave = tid >> 5;
  const int lane = tid & 31;

  for (int idx = tid; idx < 16 * HID; idx += 128) {
    sA[idx] = ah[(long)row0 * HID + idx];
    sH[idx] = h [(long)row0 * HID + idx];
  }
  __syncthreads();

  const int m    = lane & 15;
  const int kh   = (lane >> 4) * 2;
  const int half = lane >> 4;
  const int n    = wave * 16 + (lane & 15);

  v8f c = {};
  #pragma unroll 4
  for (int k = 0; k < HID; k += 4) {
    v2f a, b;
    a.x = sA[m * HID + k + kh];
    a.y = sA[m * HID + k + kh + 1];
    b.x = Wl[n * HID + k + kh];
    b.y = Wl[n * HID + k + kh + 1];
    c = __builtin_amdgcn_wmma_f32_16x16x4_f32(false, a, false, b,
                                              (short)0, c, false, false);
    a.x = sH[m * HID + k + kh];
    a.y = sH[m * HID + k + kh + 1];
    b.x = Wr[n * HID + k + kh];
    b.y = Wr[n * HID + k + kh + 1];
    c = __builtin_amdgcn_wmma_f32_16x16x4_f32(false, a, false, b,
                                              (short)0, c, false, false);
  }

  const float bias = bl[n] + br[n];
  float t[8];

  // Row max (over this wave's 16 cols), then across the 4 waves via LDS.
  #pragma unroll
  for (int r = 0; r < 8; ++r) {
    t[r] = c[r] + bias;
    float mx = t[r];
    mx = fmaxf(mx, __shfl_xor(mx, 1));
    mx = fmaxf(mx, __shfl_xor(mx, 2));
    mx = fmaxf(mx, __shfl_xor(mx, 4));
    mx = fmaxf(mx, __shfl_xor(mx, 8));
    if ((lane & 15) == 0) pmax[wave][r + half * 8] = mx;
  }
  __syncthreads();

  float rmax[8];
  #pragma unroll
  for (int r = 0; r < 8; ++r) {
    const int rowm = r + half * 8;
    rmax[r] = fmaxf(fmaxf(pmax[0][rowm], pmax[1][rowm]),
                    fmaxf(pmax[2][rowm], pmax[3][rowm]));
    float s = expf(t[r] - rmax[r]);
    s += __shfl_xor(s, 1);
    s += __shfl_xor(s, 2);
    s += __shfl_xor(s, 4);
    s += __shfl_xor(s, 8);
    if ((lane & 15) == 0) psum[wave][rowm] = s;
  }
  __syncthreads();

  #pragma unroll
  for (int r = 0; r < 8; ++r) {
    const int rowm = r + half * 8;
    const float ssum = psum[0][rowm] + psum[1][rowm] +
                       psum[2][rowm] + psum[3][rowm];
    out[(long)(row0 + rowm) * NCLS + n] = t[r] - rmax[r] - logf(ssum);
  }
}

// ---------------------------------------------------------------------------
// Host launcher
// ---------------------------------------------------------------------------
static inline size_t align_up(size_t v, size_t a) { return (v + a - 1) & ~(a - 1); }

extern "C" void kernel_launch(void* const* d_in, const int* in_sizes, int n_in,
                              void* d_out, int out_size, void* d_ws, size_t ws_size,
                              hipStream_t stream) {
  (void)in_sizes; (void)n_in; (void)out_size; (void)ws_size;

  const float* x    = (const float*)d_in[0];   // [N, F_IN]
  const float* ev   = (const float*)d_in[1];   // [E]
  const float* Wl1  = (const float*)d_in[2];   // [HID, F_IN]
  const float* bl1  = (const float*)d_in[3];   // [HID]
  const float* Wr1  = (const float*)d_in[4];   // [HID, F_IN]
  const float* br1  = (const float*)d_in[5];   // [HID]
  const float* Wl2  = (const float*)d_in[6];   // [NCLS, HID]
  const float* bl2  = (const float*)d_in[7];   // [NCLS]
  const float* Wr2  = (const float*)d_in[8];   // [NCLS, HID]
  const float* br2  = (const float*)d_in[9];   // [NCLS]
  const int*   erow = (const int*)d_in[10];    // [E] sorted
  const int*   ecol = (const int*)d_in[11];    // [E]
  float* out = (float*)d_out;                  // [N, NCLS]

  // Workspace carve-up (~143 MB)
  char* ws = (char*)d_ws;
  size_t off = 0;
  int* row_ptr = (int*)(ws + off);
  off = align_up(off + (size_t)(N_NODES + 1) * sizeof(int), 256);
  float* ax = (float*)(ws + off);
  off = align_up(off + (size_t)N_NODES * F_IN * sizeof(float), 256);
  float* hbuf = (float*)(ws + off);
  off = align_up(off + (size_t)N_NODES * HID * sizeof(float), 256);
  float* ah = (float*)(ws + off);
  off = align_up(off + (size_t)N_NODES * HID * sizeof(float), 256);

  // 1) CSR row pointers
  build_row_ptr_kernel<<<(N_NODES + 256) / 256, 256, 0, stream>>>(erow, row_ptr);
  // 2) SpMM on x
  spmm_csr_kernel<F_IN><<<N_NODES, 128, 0, stream>>>(row_ptr, ecol, ev, x, ax);
  // 3) Layer 1 (WMMA + fused l1norm + ReLU)
  layer1_kernel<<<N_NODES / 16, 256, 0, stream>>>(ax, x, Wl1, bl1, Wr1, br1, hbuf);
  // 4) SpMM on h
  spmm_csr_kernel<HID><<<N_NODES, 128, 0, stream>>>(row_ptr, ecol, ev, hbuf, ah);
  // 5) Layer 2 (WMMA + fused log_softmax)
  layer2_kernel<<<N_NODES / 16, 128, 0, stream>>>(ah, hbuf, Wl2, bl2, Wr2, br2, out);
}